// _apply_aconv_58901181497896
// MI455X (gfx1250) — compile-verified
//
#include <hip/hip_runtime.h>

// Fused kernel for:
//   logits = einsum('btsd,df->btsf', dconv, Ww)        (bw cancels in softmax)
//   aconv  = softmax(logits, axis=s)
//   prod   = einsum('btsf,bsh->bthf', aconv, input_seq)
//   out    = prod.reshape(b,t,H*F) @ Wc + bc
//
// B=2, T=S=512, H=64, F=32, D=64. One workgroup per (b,t); 8 wave32s.
// GEMMs use V_WMMA_F32_16X16X4_F32 (fp32, matches reference precision).
// dconv (128 MB total) is streamed through LDS with double-buffered
// GLOBAL_LOAD_ASYNC_TO_LDS_B128 (ASYNCcnt pipeline) so the HBM stream
// overlaps WMMA compute. input_seq[b] (128 KB, shared by 512 blocks) is
// L2-resident and read directly by GEMM2's A-fragments.

typedef __attribute__((ext_vector_type(2))) float v2f;
typedef __attribute__((ext_vector_type(4))) float v4f;
typedef __attribute__((ext_vector_type(8))) float v8f;

#define BB 2
#define TT 512
#define SS 512
#define HH 64
#define FF 32
#define DD 64
#define SC 64            // s-chunk rows
#define NCHUNK (SS / SC) // 8
#define CHUNK_BYTES (SC * DD * 4)                 // 16384
#define ASYNC_ROUNDS (CHUNK_BYTES / (256 * 16))   // 4 b128 ops per thread per chunk

__device__ __forceinline__ void async_copy_b128(unsigned lds_off, const void* gaddr) {
    // vdst = LDS byte address VGPR, vaddr = 64-bit global address VGPR pair
    asm volatile("global_load_async_to_lds_b128 %0, %1, off"
                 :: "v"(lds_off), "v"(gaddr)
                 : "memory");
}

__global__ __launch_bounds__(256)
void fused_aconv_kernel(const float* __restrict__ input_seq,
                        const float* __restrict__ dconv,
                        const float* __restrict__ Ww,
                        const float* __restrict__ Wc,
                        const float* __restrict__ bc,
                        float* __restrict__ out)
{
    __shared__ float lds_ww[DD * FF];          // 8 KB   Ww staged
    __shared__ float lds_dconv[2][SC * DD];    // 32 KB  dconv ping-pong (async DMA target)
    __shared__ float lds_e[SC * FF];           // 8 KB   exp(logits) [s][f] (reused as GEMV scratch)
    __shared__ float lds_prod[HH * FF];        // 8 KB   product [h][f]
    __shared__ float lds_colsum[FF];           // softmax denominators

    const int tid   = threadIdx.x;
    const int wave  = tid >> 5;                // wave32
    const int lane  = tid & 31;
    const int lhalf = lane >> 4;               // K-pair selector in A/B layouts
    const int l16   = lane & 15;               // M (A) / N (B,C,D) index within tile

    const int bt = blockIdx.x;
    const int b  = bt / TT;

    const int mtile = wave & 3;                // GEMM1: s-subtile; GEMM2: h-tile
    const int ntile = wave >> 2;               // f-tile (0..1)
    const int f0 = ntile * 16;
    const int h0 = mtile * 16;

    const float* dconv_bt = dconv + (size_t)bt * SS * DD;
    const float* in_b     = input_seq + (size_t)b * SS * HH;

    // LDS byte offset of the ping-pong buffer (generic ptr low 32 bits == LDS offset)
    const unsigned lds_dc_base = (unsigned)(size_t)(&lds_dconv[0][0]);

    // ---- prologue: kick off async DMA of chunk 0 into buffer 0 ----
    {
        const char* g = (const char*)dconv_bt;
        #pragma unroll
        for (int r = 0; r < ASYNC_ROUNDS; ++r)
            async_copy_b128(lds_dc_base + tid * 16 + r * 4096, g + tid * 16 + r * 4096);
    }

    // ---- stage Ww (2048 floats, contiguous) while DMA runs ----
    {
        const v4f* src = (const v4f*)Ww;
        v4f*       dst = (v4f*)lds_ww;
        #pragma unroll
        for (int i = 0; i < (DD * FF / 4) / 256; ++i)
            dst[tid + i * 256] = src[tid + i * 256];
    }
    if (tid < FF) lds_colsum[tid] = 0.0f;
    __syncthreads();

    // ---- prebuild GEMM1 B-fragments from Ww ([d][f] row-major) ----
    // B 4x16 layout: v.x = B[kb][col], v.y = B[kb+1][col], kb = 4*kk + 2*lhalf
    v2f bfrag[16];
    #pragma unroll
    for (int kk = 0; kk < 16; ++kk) {
        const int kb = 4 * kk + 2 * lhalf;
        bfrag[kk].x = lds_ww[kb * FF + f0 + l16];
        bfrag[kk].y = lds_ww[(kb + 1) * FF + f0 + l16];
    }

    v8f   acc2   = {};   // GEMM2 accumulator (persistent across chunks)
    float colacc = 0.f;  // per-lane partial softmax denominator for col f0+l16

    for (int c = 0; c < NCHUNK; ++c) {
        const int s0 = c * SC;
        const float* buf = lds_dconv[c & 1];

        if (c + 1 < NCHUNK) {
            // issue async DMA of chunk c+1 into the other buffer
            // (end-of-iteration barrier of chunk c-1 guarantees it is free)
            const char* g = (const char*)(dconv_bt + (size_t)(c + 1) * SC * DD);
            const unsigned lbase = lds_dc_base + ((c + 1) & 1) * CHUNK_BYTES;
            #pragma unroll
            for (int r = 0; r < ASYNC_ROUNDS; ++r)
                async_copy_b128(lbase + tid * 16 + r * 4096, g + tid * 16 + r * 4096);
            if (c + 2 < NCHUNK)  // warm L2 one more chunk ahead
                __builtin_prefetch(dconv_bt + (size_t)(c + 2) * SC * DD + tid * 16, 0, 0);
            // async loads complete in order: <=4 outstanding => chunk c landed
            asm volatile("s_wait_asynccnt 4" ::: "memory");
        } else {
            asm volatile("s_wait_asynccnt 0" ::: "memory");
        }
        __syncthreads();

        // ---- GEMM1: logits tile [16s x 16f], K = D = 64 in 16 steps of 4 ----
        v8f acc1 = {};
        #pragma unroll
        for (int kk = 0; kk < 16; ++kk) {
            const int kb = 4 * kk + 2 * lhalf;
            // A 16x4 layout: lane holds M=l16, K pair (kb,kb+1) -> ds_load_b64
            v2f a = *(const v2f*)&buf[(mtile * 16 + l16) * DD + kb];
            acc1 = __builtin_amdgcn_wmma_f32_16x16x4_f32(
                false, a, false, bfrag[kk], (short)0, acc1, false, false);
        }

        // ---- exp + column partial sums; spill e to LDS [s][f] ----
        // C layout: element r of lane L -> row mtile*16 + r + 8*lhalf, col f0+l16
        #pragma unroll
        for (int r = 0; r < 8; ++r) {
            const float e = __expf(acc1[r]);
            colacc += e;
            const int srow = mtile * 16 + r + 8 * lhalf;
            lds_e[srow * FF + f0 + l16] = e;
        }
        __syncthreads();

        // ---- GEMM2 partial: product[h,f] += sum_s in[s,h] * e[s,f] ----
        // A[h,s] = input_seq[b,s,h] read straight from global (L2-resident)
        #pragma unroll
        for (int kk = 0; kk < 16; ++kk) {
            const int kb = 4 * kk + 2 * lhalf;
            v2f a, bb;
            a.x  = in_b[(size_t)(s0 + kb) * HH + h0 + l16];
            a.y  = in_b[(size_t)(s0 + kb + 1) * HH + h0 + l16];
            bb.x = lds_e[kb * FF + f0 + l16];
            bb.y = lds_e[(kb + 1) * FF + f0 + l16];
            acc2 = __builtin_amdgcn_wmma_f32_16x16x4_f32(
                false, a, false, bb, (short)0, acc2, false, false);
        }
        __syncthreads();  // all waves done with lds_e and buf[c&1]
    }

    // ---- finalize softmax denominators across waves ----
    atomicAdd(&lds_colsum[f0 + l16], colacc);   // ds_add_f32
    __syncthreads();

    // ---- normalize and write product tile to LDS [h][f] ----
    {
        const float inv = 1.0f / lds_colsum[f0 + l16];
        #pragma unroll
        for (int r = 0; r < 8; ++r) {
            const int h = h0 + r + 8 * lhalf;
            lds_prod[h * FF + f0 + l16] = acc2[r] * inv;
        }
    }
    __syncthreads();

    // ---- GEMV: out[h'] = flatten(prod) . Wc[:, h'] + bc[h'] ----
    // flat index i = h*F + f matches reshape(b,t,H*F); Wc is [H*F, H] row-major.
    {
        const int hp   = tid & 63;
        const int part = tid >> 6;  // 0..3, each sums 512 of 2048 terms
        float s = 0.f;
        const float* wc = Wc + hp;
        for (int i = part * 512; i < part * 512 + 512; ++i)
            s += lds_prod[i] * wc[(size_t)i * HH];
        lds_e[part * 64 + hp] = s;  // reuse lds_e as reduction scratch
    }
    __syncthreads();
    if (tid < 64) {
        const float s = lds_e[tid] + lds_e[64 + tid] + lds_e[128 + tid] +
                        lds_e[192 + tid] + bc[tid];
        out[(size_t)bt * HH + tid] = s;
    }
}

extern "C" void kernel_launch(void* const* d_in, const int* in_sizes, int n_in,
                              void* d_out, int out_size, void* d_ws, size_t ws_size,
                              hipStream_t stream) {
    const float* input_seq = (const float*)d_in[0];  // [B,T,H]
    const float* dconv     = (const float*)d_in[1];  // [B,T,T,D]
    const float* Ww        = (const float*)d_in[2];  // [D,F]
    // d_in[3] = bw: constant along the softmax axis -> cancels, unused
    const float* Wc        = (const float*)d_in[4];  // [H*F,H]
    const float* bc        = (const float*)d_in[5];  // [H]
    float* out = (float*)d_out;

    fused_aconv_kernel<<<dim3(BB * TT), dim3(256), 0, stream>>>(
        input_seq, dconv, Ww, Wc, bc, out);
}